// PreQ_5153960755940
// MI455X (gfx1250) — compile-verified
//
#include <hip/hip_runtime.h>
#include <math.h>

// fp32 WMMA fragment types: A/B 16x4 f32 = 2 VGPRs/lane, C/D 16x16 f32 = 8 VGPRs/lane
typedef float v2f __attribute__((ext_vector_type(2)));
typedef float v8f __attribute__((ext_vector_type(8)));

#define NQ     10
#define BATCH  4096
#define MTILE  16     // batch rows per block
#define NWAVES 8      // 256 threads / wave32

__device__ __forceinline__ int imin(int a, int b) { return a < b ? a : b; }

// One fused-GEMM layer: actOut[h][16][ldOut] = relu(actIn[h][16][ldIn] @ W[K,N] + bias)
// Uses V_WMMA_F32_16X16X4_F32. Lane layout (ISA 7.12.2, 32-bit A 16x4):
//   lanes 0-15: M=lane, K=k..k+1 ; lanes 16-31: M=lane-16, K=k+2..k+3
// B 4x16 analogous: lanes 0-15 rows k,k+1 ; lanes 16-31 rows k+2,k+3 ; col = lane&15.
//
// All global loads are made unconditionally in-bounds by clamping indices; the
// resulting garbage columns/rows either multiply by zero-padded A entries or
// land in D columns that are stored as 0.  -> no EXEC divergence in the hot loop.
__device__ __forceinline__ void layer_wmma(
    const float* __restrict__ W, const float* __restrict__ bias,
    const float* actIn, float* actOut,
    int K, int N, int ldIn, int ldOut, int nTiles, bool relu, int tid)
{
  const int lane  = tid & 31;
  const int wave  = tid >> 5;
  const int colT  = lane & 15;          // column within 16-wide N tile
  const int koff  = (lane >= 16) ? 2 : 0;
  const int rowA  = lane & 15;          // A-matrix row for this lane
  const int rbase = (lane >= 16) ? 8 : 0;
  const int nJobs = 2 * nTiles;         // (half h) x (N tile)

  for (int job = wave; job < nJobs; job += NWAVES) {
    const int h    = job / nTiles;
    const int n    = job - h * nTiles;
    const int col  = n * 16 + colT;
    const bool colOK = (col < N);
    const int colc = imin(col, N - 1);           // clamped -> always in-bounds
    const float* aBase = actIn + (h * MTILE + rowA) * ldIn + koff;
    const float* wCol  = W + colc;               // column base pointer

    v8f acc = {0.f, 0.f, 0.f, 0.f, 0.f, 0.f, 0.f, 0.f};

    const int kFull = K & ~3;
    for (int k = 0; k < kFull; k += 4) {
      v2f a; a.x = aBase[k]; a.y = aBase[k + 1];          // ds_load_b64
      const int kb = k + koff;
      v2f b; b.x = wCol[kb * N]; b.y = wCol[(kb + 1) * N]; // 2x global_load_b32
      acc = __builtin_amdgcn_wmma_f32_16x16x4_f32(
          /*neg_a=*/false, a, /*neg_b=*/false, b,
          /*c_mod=*/(short)0, acc, /*reuse_a=*/false, /*reuse_b=*/false);
    }
    if (kFull < K) {                      // K=50 tail: A zero-padded past K
      const int k  = kFull;
      const int kb = k + koff;
      v2f a; a.x = aBase[k]; a.y = aBase[k + 1];
      v2f b;
      b.x = wCol[imin(kb,     K - 1) * N];   // clamped row: A side is 0 there
      b.y = wCol[imin(kb + 1, K - 1) * N];
      acc = __builtin_amdgcn_wmma_f32_16x16x4_f32(
          false, a, false, b, (short)0, acc, false, false);
    }

    const float bv = bias[colc];
#pragma unroll
    for (int r = 0; r < 8; ++r) {
      const int row = rbase + r;                          // D: VGPR r -> M=row
      float v = acc[r] + bv;
      if (relu) v = fmaxf(v, 0.f);
      // zero-fill pad columns so next layer's K loop can read them safely
      actOut[(h * MTILE + row) * ldOut + col] = colOK ? v : 0.f;
    }
  }
}

// LDS plan (floats), total 16256 f32 = 65,024 B:
//   A    @     0 : 2*16*300 = 9600   (x tile, both halves)
//   act1 @  9600 : 2*16*208 = 6656   (layer1 out)
//   act2 @     0 : 2*16*112 = 3584   (aliases dead A)
//   act3 @  3584 : 2*16* 64 = 2048
//   pbuf @  5632 : 2*16* 32 = 1024   (MLP params p1/p2)
__global__ __launch_bounds__(256) void preq_fused_kernel(
    const float* __restrict__ x,
    const float* __restrict__ W1, const float* __restrict__ b1,
    const float* __restrict__ W2, const float* __restrict__ b2,
    const float* __restrict__ W3, const float* __restrict__ b3,
    const float* __restrict__ W4, const float* __restrict__ b4,
    float* __restrict__ out)
{
  __shared__ float smem[16256];
  float* A    = smem;
  float* act1 = smem + 9600;
  float* act2 = smem;
  float* act3 = smem + 3584;
  float* pbuf = smem + 5632;

  const int tid = threadIdx.x;
  const int b0  = blockIdx.x * MTILE;

  // Stage x tile: A[h][row][0..299] = x[b0+row][h*300 + c]  (coalesced over c)
  for (int i = tid; i < 2 * MTILE * 300; i += 256) {
    const int h = i / (MTILE * 300);
    const int r = (i / 300) % MTILE;
    const int c = i % 300;
    A[(h * MTILE + r) * 300 + c] = x[(b0 + r) * 600 + h * 300 + c];
  }
  __syncthreads();

  layer_wmma(W1, b1, A,    act1, 300, 200, 300, 208, 13, true,  tid);
  __syncthreads();
  layer_wmma(W2, b2, act1, act2, 200, 100, 208, 112,  7, true,  tid);
  __syncthreads();
  layer_wmma(W3, b3, act2, act3, 100,  50, 112,  64,  4, true,  tid);
  __syncthreads();
  layer_wmma(W4, b4, act3, pbuf,  50,  29,  64,  32,  2, false, tid);
  __syncthreads();

  // Closed-form fidelity: CRz chain on |0..0> is identity; state is a product
  // state  ⊗_q [cos(a_q/2) e^{-i z_q/2}, sin(a_q/2) e^{+i z_q/2}]  so
  // |<s2|s1>|^2 = prod_q [cos^2(d) cos^2((a1-a2)/2) + sin^2(d) cos^2((a1+a2)/2)]
  if (tid < MTILE) {
    const float* p1 = pbuf + tid * 32;              // half 0
    const float* p2 = pbuf + (MTILE + tid) * 32;    // half 1
    float fid = 1.f;
#pragma unroll
    for (int q = 0; q < NQ; ++q) {
      const int zi = (q == 0) ? 1 : q;              // faithful rx_idx quirk
      const float a1 = p1[NQ + q], a2 = p2[NQ + q];
      const float d  = 0.5f * (p1[zi] - p2[zi]);
      const float cm = cosf(0.5f * (a1 - a2));
      const float cp = cosf(0.5f * (a1 + a2));
      const float cd = cosf(d), sd = sinf(d);
      fid *= cd * cd * cm * cm + sd * sd * cp * cp;
    }
    out[b0 + tid] = fid;
  }
}

extern "C" void kernel_launch(void* const* d_in, const int* in_sizes, int n_in,
                              void* d_out, int out_size, void* d_ws, size_t ws_size,
                              hipStream_t stream) {
  (void)in_sizes; (void)n_in; (void)out_size; (void)d_ws; (void)ws_size;
  const float* x  = (const float*)d_in[0];
  const float* W1 = (const float*)d_in[1];
  const float* b1 = (const float*)d_in[2];
  const float* W2 = (const float*)d_in[3];
  const float* b2 = (const float*)d_in[4];
  const float* W3 = (const float*)d_in[5];
  const float* b3 = (const float*)d_in[6];
  const float* W4 = (const float*)d_in[7];
  const float* b4 = (const float*)d_in[8];
  float* out = (float*)d_out;

  preq_fused_kernel<<<BATCH / MTILE, 256, 0, stream>>>(
      x, W1, b1, W2, b2, W3, b3, W4, b4, out);
}